// E8P12_codebook_9740985828125
// MI455X (gfx1250) — compile-verified
//
#include <hip/hip_runtime.h>

typedef float v2f __attribute__((ext_vector_type(2)));
typedef float v8f __attribute__((ext_vector_type(8)));

#define WAVES 8
#define TPB 256
#define ROWS_PER_BLOCK 128   // 8 waves * 16 rows
#define TILES_PER_CHUNK 64   // 64 tiles * 16 centroids * 8 floats = 32KB LDS

__global__ __launch_bounds__(TPB)
void e8p_quant_kernel(const float* __restrict__ X,
                      const float* __restrict__ gridPart,
                      const float* __restrict__ gridNorm,
                      const int*   __restrict__ partAbsMap,
                      float* __restrict__ outVals,
                      int*   __restrict__ outIdx,
                      int rows, int G, int ntiles)
{
    __shared__ float ldsGrid[TILES_PER_CHUNK * 16 * 8];
    __shared__ float ldsNorm[TILES_PER_CHUNK * 16];
    __shared__ int   ldsBest[WAVES * 2 * 16];

    const int tid  = threadIdx.x;
    const int wave = tid >> 5;
    const int lane = tid & 31;
    const int half = lane >> 4;     // wave32: two 16-lane halves
    const int ln   = lane & 15;
    const int k0   = half * 2;      // A/B K-pair owned by this half (ISA 7.12.2)

    const int rowBase = blockIdx.x * ROWS_PER_BLOCK + wave * 16;
    const int myRow   = rowBase + ln;

    // Each lane holds the full 8-float row (lanes l and l+16 share row l&15).
    float x[8];
    if (myRow < rows) {
        const float4* xp = (const float4*)(X + (size_t)myRow * 8);
        float4 lo = xp[0], hi = xp[1];
        x[0]=lo.x; x[1]=lo.y; x[2]=lo.z; x[3]=lo.w;
        x[4]=hi.x; x[5]=hi.y; x[6]=hi.z; x[7]=hi.w;
    } else {
        #pragma unroll
        for (int i = 0; i < 8; ++i) x[i] = 0.f;
    }

    // Build WMMA A fragments for both shift variants: A = 2 * X_part
    // (factor 2 of the score 2*X@g^T - ||g||^2 folded into A).
    v2f a0[2], a1[2];
    #pragma unroll
    for (int s = 0; s < 2; ++s) {
        float q = (s == 0) ? 0.25f : -0.25f;
        float xp8[8]; int negc = 0;
        #pragma unroll
        for (int i = 0; i < 8; ++i) {
            float y = x[i] + q;
            if (y < 0.f) negc++;
            xp8[i] = fabsf(y);
        }
        if (negc & 1) xp8[7] = -xp8[7];   // odd rows: negate col 7
        a0[s].x = 2.f * xp8[k0];     a0[s].y = 2.f * xp8[k0 + 1];
        a1[s].x = 2.f * xp8[k0 + 4]; a1[s].y = 2.f * xp8[k0 + 5];
    }

    // Running per-lane argmax: VGPR v <-> row (v + half*8); lane column = ln.
    float bestS[2][8];
    int   bestI[2][8];
    #pragma unroll
    for (int s = 0; s < 2; ++s)
        #pragma unroll
        for (int v = 0; v < 8; ++v) { bestS[s][v] = -3.0e38f; bestI[s][v] = 0; }

    for (int t0 = 0; t0 < ntiles; t0 += TILES_PER_CHUNK) {
        const int tc   = min(TILES_PER_CHUNK, ntiles - t0);
        const int base = t0 * 16;
        const int nc   = tc * 16;
        __syncthreads();  // previous chunk fully consumed
        for (int i = tid; i < nc * 8; i += TPB) {
            int gi = base * 8 + i;
            ldsGrid[i] = (gi < G * 8) ? gridPart[gi] : 0.f;
        }
        for (int n = tid; n < nc; n += TPB) {
            int gn = base + n;
            ldsNorm[n] = (gn < G) ? gridNorm[gn] : 1.0e30f;  // padded never wins
        }
        __syncthreads();

        for (int t = 0; t < tc; ++t) {
            const float* g = ldsGrid + (t * 16 + ln) * 8;
            float4 glo = *(const float4*)g;
            float4 ghi = *(const float4*)(g + 4);
            // B[k][n] = grid[tile*16+n][k]; same K permutation as A.
            v2f b0, b1;
            if (half) { b0.x = glo.z; b0.y = glo.w; b1.x = ghi.z; b1.y = ghi.w; }
            else      { b0.x = glo.x; b0.y = glo.y; b1.x = ghi.x; b1.y = ghi.y; }
            float nrm = ldsNorm[t * 16 + ln];
            int  cIdx = base + t * 16 + ln;

            // Fold the -||g||^2 term into the WMMA accumulator:
            // C[m][n] = -norm[n] (same splat feeds both variants' chains).
            v8f cinit;
            #pragma unroll
            for (int v = 0; v < 8; ++v) cinit[v] = -nrm;

            #pragma unroll
            for (int s = 0; s < 2; ++s) {
                // D = A0*B0 + A1*B1 - norm  (K=8 via two chained K=4 WMMAs)
                v8f c;
                c = __builtin_amdgcn_wmma_f32_16x16x4_f32(false, a1[s], false, b1,
                                                          (short)0, cinit, false, false);
                c = __builtin_amdgcn_wmma_f32_16x16x4_f32(false, a0[s], false, b0,
                                                          (short)0, c, false, false);
                #pragma unroll
                for (int v = 0; v < 8; ++v) {
                    float sc = c[v];
                    if (sc > bestS[s][v]) { bestS[s][v] = sc; bestI[s][v] = cIdx; }
                }
            }
        }
    }

    // Cross-lane argmax butterfly within each 16-lane half (xor 1,2,4,8 stays in-half).
    #pragma unroll
    for (int s = 0; s < 2; ++s) {
        #pragma unroll
        for (int v = 0; v < 8; ++v) {
            float bs = bestS[s][v];
            int   bi = bestI[s][v];
            #pragma unroll
            for (int m = 1; m < 16; m <<= 1) {
                float os = __shfl_xor(bs, m, 32);
                int   oi = __shfl_xor(bi, m, 32);
                if (os > bs || (os == bs && oi < bi)) { bs = os; bi = oi; }
            }
            bestI[s][v] = bi;
        }
    }
    if (ln == 0) {  // lanes 0 (rows 0..7) and 16 (rows 8..15)
        #pragma unroll
        for (int s = 0; s < 2; ++s)
            #pragma unroll
            for (int v = 0; v < 8; ++v)
                ldsBest[(wave * 2 + s) * 16 + half * 8 + v] = bestI[s][v];
    }
    __syncthreads();

    // Epilogue: lane r (0..15) owns row r.
    if (half == 0 && myRow < rows) {
        float vals2[2][8];
        float err2[2];
        int   code[2];
        #pragma unroll
        for (int s = 0; s < 2; ++s) {
            int bi = ldsBest[(wave * 2 + s) * 16 + ln];
            const float4* gp = (const float4*)(gridPart + (size_t)bi * 8);
            float4 glo = gp[0], ghi = gp[1];
            float g[8] = {glo.x, glo.y, glo.z, glo.w, ghi.x, ghi.y, ghi.z, ghi.w};
            float q = (s == 0) ? 0.25f : -0.25f;
            float y[8], mask[8]; int negc = 0;
            #pragma unroll
            for (int i = 0; i < 8; ++i) {
                y[i] = x[i] + q;
                bool n = y[i] < 0.f;
                negc += n ? 1 : 0;
                mask[i] = n ? -1.f : 1.f;
            }
            if (negc & 1) mask[7] = -mask[7];
            float e = 0.f, sabs = 0.f;
            bool sb[8];
            #pragma unroll
            for (int i = 0; i < 8; ++i) {
                float v = g[i] * mask[i];
                vals2[s][i] = v;
                float d = y[i] - v;
                e += d * d;
                sabs += fabsf(g[i]);
                sb[i] = (g[i] < 0.f) != (mask[i] < 0.f);
            }
            err2[s] = e;
            int absIdx = partAbsMap[bi];
            // grid_abs_odd[absIdx] == parity of sum(|abs_grid row|) == parity of sum(|g|)
            bool gao = (((int)rintf(sabs)) & 1) != 0;
            const int PERM[8] = {0, 2, 4, 6, 1, 3, 5, 7};
            bool sm[8];
            #pragma unroll
            for (int j = 0; j < 8; ++j) sm[j] = sb[PERM[j]];
            sm[7] = sm[7] != gao;
            sm[0] = sm[0] != (s == 0);   // parity bit: True for plus variant
            int mi = 0;
            #pragma unroll
            for (int j = 0; j < 8; ++j) mi |= ((int)sm[j]) << j;
            code[s] = (absIdx << 8) + mi;
        }
        bool which = err2[0] < err2[1];  // sqrt is monotone: compare squares
        float outr[8];
        #pragma unroll
        for (int i = 0; i < 8; ++i)
            outr[i] = which ? (vals2[0][i] - 0.25f) : (vals2[1][i] + 0.25f);
        float4* op = (float4*)(outVals + (size_t)myRow * 8);
        op[0] = make_float4(outr[0], outr[1], outr[2], outr[3]);
        op[1] = make_float4(outr[4], outr[5], outr[6], outr[7]);
        outIdx[myRow] = which ? code[0] : code[1];
    }
}

extern "C" void kernel_launch(void* const* d_in, const int* in_sizes, int n_in,
                              void* d_out, int out_size, void* d_ws, size_t ws_size,
                              hipStream_t stream) {
    const float* X          = (const float*)d_in[0];
    const float* gridPart   = (const float*)d_in[1];
    const float* gridNorm   = (const float*)d_in[2];
    const int*   partAbsMap = (const int*)d_in[3];
    // d_in[4] = grid_abs_odd (recomputed in-kernel), d_in[5] = bit_map (2^j) -- unused

    const int rows   = in_sizes[0] / 8;
    const int G      = in_sizes[1] / 8;
    const int ntiles = (G + 15) / 16;

    float* outVals = (float*)d_out;
    int*   outIdx  = (int*)((float*)d_out + (size_t)rows * 8);

    const int blocks = (rows + ROWS_PER_BLOCK - 1) / ROWS_PER_BLOCK;
    hipLaunchKernelGGL(e8p_quant_kernel, dim3(blocks), dim3(TPB), 0, stream,
                       X, gridPart, gridNorm, partAbsMap, outVals, outIdx,
                       rows, G, ntiles);
}